// TensorReduce_88536455839845
// MI455X (gfx1250) — compile-verified
//
#include <hip/hip_runtime.h>

// Volume-rendering composite (hyperreel TensorReduce) for MI455X / gfx1250.
// Bandwidth-bound: ~252 MB streamed once -> ~11 us at 23.3 TB/s.
// One wave32 per ray; exclusive cumprod via DPP prefix-product scan (VALU,
// no LDS crossbar); rgba double-buffered through LDS with gfx1250 async
// global->LDS loads (ASYNCcnt), non-temporal since the stream exceeds L2.

#define S_SAMPLES 192
#define CHUNK 32
#define NCHUNK (S_SAMPLES / CHUNK)      // 6
#define WAVES_PER_BLOCK 8
#define THREADS (WAVES_PER_BLOCK * 32)  // 256

typedef int v4i __attribute__((ext_vector_type(4)));
typedef __attribute__((address_space(1))) v4i g_v4i;   // global
typedef __attribute__((address_space(3))) v4i l_v4i;   // LDS

__device__ __forceinline__ void async_copy_b128_nt(const void* g, void* l) {
#if __has_builtin(__builtin_amdgcn_global_load_async_to_lds_b128)
  // (v4i32 AS1*, v4i32 AS3*, imm offset, imm cpol); cpol=1 -> th:TH_LOAD_NT
  __builtin_amdgcn_global_load_async_to_lds_b128(
      (g_v4i*)(v4i*)(void*)g, (l_v4i*)(v4i*)l, /*offset=*/0, /*cpol=*/1);
#else
  unsigned laddr =
      (unsigned)(unsigned long long)(__attribute__((address_space(3))) char*)l;
  unsigned long long gaddr = (unsigned long long)g;
  asm volatile("global_load_async_to_lds_b128 %0, %1, off"
               :: "v"(laddr), "v"(gaddr) : "memory");
#endif
}

#if __has_builtin(__builtin_amdgcn_s_wait_asynccnt)
#define WAIT_ASYNC(n)                          \
  do {                                         \
    __builtin_amdgcn_s_wait_asynccnt(n);       \
    __asm__ __volatile__("" ::: "memory");     \
  } while (0)
#else
#define WAIT_ASYNC(n) \
  __asm__ __volatile__("s_wait_asynccnt %0" :: "n"(n) : "memory")
#endif

// x_shifted[lane] = x[lane - N] within each 16-lane row; invalid lanes -> id.
__device__ __forceinline__ float dpp_shr(float x, float id, int ctrl_is_1_2_4_8) {
  // caller passes literal ctrl; wrapped by macro below to keep immediates
  return 0.0f;  // (unused; see DPP_SHR macro)
}
#define DPP_SHR(x, id, ctrl)                                              \
  __builtin_bit_cast(float, __builtin_amdgcn_update_dpp(                  \
      __builtin_bit_cast(int, (id)), __builtin_bit_cast(int, (x)),        \
      (ctrl), 0xF, 0xF, false))

__device__ __forceinline__ float lane_read(float x, int lane) {
  return __builtin_bit_cast(
      float, __builtin_amdgcn_readlane(__builtin_bit_cast(int, x), lane));
}

__global__ __launch_bounds__(THREADS) void TensorReduce_composite_kernel(
    const float* __restrict__ rgba,   // [N, S, 4]
    const float* __restrict__ dist,   // [N, S]
    float* __restrict__ out,          // [N, 3]
    int nRays) {
  __shared__ float4 smem[WAVES_PER_BLOCK * 2 * CHUNK];  // 8 KB: dbl-buffered

  const int lane = threadIdx.x & 31;
  const int wave = threadIdx.x >> 5;
  const int ray  = blockIdx.x * WAVES_PER_BLOCK + wave;
  if (ray >= nRays) return;

  const float* rayRgba = rgba + (size_t)ray * S_SAMPLES * 4;
  const float* rayDist = dist + (size_t)ray * S_SAMPLES;
  float4* buf = smem + wave * 2 * CHUNK;

  // Kick off chunk 0 (each lane fetches its 16B sample into LDS).
  async_copy_b128_nt(rayRgba + (size_t)lane * 4, buf + lane);

  float carry = 1.0f;                    // product of (1-alpha+eps) so far
  float ax = 0.0f, ay = 0.0f, az = 0.0f;

#pragma unroll
  for (int c = 0; c < NCHUNK; ++c) {
    if (c + 1 < NCHUNK) {
      async_copy_b128_nt(rayRgba + (size_t)((c + 1) * CHUNK + lane) * 4,
                         buf + ((c + 1) & 1) * CHUNK + lane);
      WAIT_ASYNC(1);  // chunk c done; chunk c+1 still in flight
    } else {
      WAIT_ASYNC(0);
    }

    const int s = c * CHUNK + lane;
    float d0 = __builtin_nontemporal_load(rayDist + s);
    int s1 = (s + 1 < S_SAMPLES) ? (s + 1) : (S_SAMPLES - 1);
    float d1 = __builtin_nontemporal_load(rayDist + s1);
    float delta = (s == S_SAMPLES - 1) ? 1.0e4f : fabsf(d1 - d0);

    float4 v = buf[(c & 1) * CHUNK + lane];  // ds_load_b128

    float density = v.w;
    density = (d0 < 1.0e-5f) ? 0.0f : density;
    density = fmaxf(density, 0.0f);          // * DENSITY_FACTOR (1.0)

    float e1m = __expf(-delta * density);    // 1 - alpha
    float alpha = 1.0f - e1m;
    float t = e1m + 1.0e-8f;                 // (1 - alpha + EPS_ALPHA)

    // ---- Exclusive prefix product of t across the wave, all in DPP ----
    // Shift by 1 within rows (lanes 0 and 16 get identity 1.0).
    float p = DPP_SHR(t, 1.0f, 0x111);
    // Hillis-Steele inclusive multiply-scan of the shifted seq per 16-lane row.
    p *= DPP_SHR(p, 1.0f, 0x111);
    p *= DPP_SHR(p, 1.0f, 0x112);
    p *= DPP_SHR(p, 1.0f, 0x114);
    p *= DPP_SHR(p, 1.0f, 0x118);
    // p now = exclusive product within its 16-lane row.
    // Upper row needs the full product of lanes 0..15: (p*t) at lane 15.
    float rowfull = p * t;
    float low16 = lane_read(rowfull, 15);
    p *= (lane >= 16) ? low16 : 1.0f;        // global exclusive prefix
    float total = lane_read(p * t, 31);      // product of all 32 t's

    float trans = carry * p;
    float w = alpha * trans;

    float r = 1.0f / (1.0f + __expf(-v.x));
    float g = 1.0f / (1.0f + __expf(-v.y));
    float b = 1.0f / (1.0f + __expf(-v.z));
    ax += w * r;
    ay += w * g;
    az += w * b;

    carry *= total;
  }

  // Butterfly reduction across the wave (result replicated in all lanes).
#pragma unroll
  for (int off = 16; off >= 1; off >>= 1) {
    ax += __shfl_xor(ax, off, 32);
    ay += __shfl_xor(ay, off, 32);
    az += __shfl_xor(az, off, 32);
  }

  if (lane == 0) {
    float* o = out + (size_t)ray * 3;
    o[0] = ax;
    o[1] = ay;
    o[2] = az;
  }
}

extern "C" void kernel_launch(void* const* d_in, const int* in_sizes, int n_in,
                              void* d_out, int out_size, void* d_ws,
                              size_t ws_size, hipStream_t stream) {
  const float* rgba = (const float*)d_in[0];   // [N, S, 4] f32
  const float* dist = (const float*)d_in[1];   // [N, S]    f32
  float* out = (float*)d_out;                  // [N, 3]    f32

  int nRays = in_sizes[1] / S_SAMPLES;
  int blocks = (nRays + WAVES_PER_BLOCK - 1) / WAVES_PER_BLOCK;

  hipLaunchKernelGGL(TensorReduce_composite_kernel, dim3(blocks), dim3(THREADS),
                     0, stream, rgba, dist, out, nRays);
}